// LoTD_53077205844612
// MI455X (gfx1250) — compile-verified
//
#include <hip/hip_runtime.h>
#include <stdint.h>

typedef __attribute__((ext_vector_type(2))) float v2f;
typedef __attribute__((ext_vector_type(4))) float v4f;

#define NLVL   16
#define LOG2T  19
#define TMASK  ((1u << LOG2T) - 1u)
#define PRIME_Y 2654435761u
#define PRIME_Z 805459861u
#define BLOCK  256

// LOD_RES - 1, baked as immediates via full unroll
static constexpr float kResM1[NLVL] = {
    15.f,  21.f,  29.f,  40.f,  54.f,  74.f,  101.f,  137.f,
    187.f, 254.f, 345.f, 469.f, 637.f, 865.f, 1175.f, 1596.f};

__global__ __launch_bounds__(BLOCK)
void lotd_hashgrid_fwd(const float* __restrict__ xg,
                       const float* __restrict__ grid,
                       float* __restrict__ out, int N)
{
    __shared__ float sx[BLOCK * 3];   // 3 KB staging for this block's positions

    const int       tid = threadIdx.x;
    const long long n   = (long long)blockIdx.x * BLOCK + tid;
    const bool fullBlock = ((long long)(blockIdx.x + 1) * BLOCK) <= (long long)N;

    // ---- gfx1250 async copy: 12 KB of x -> LDS (3 x 4096B chunks, 16B/lane) ----
    if (fullBlock) {
        const uint64_t gbase  = (uint64_t)(uintptr_t)(xg + (size_t)blockIdx.x * (BLOCK * 3));
        const uint32_t voff   = (uint32_t)tid * 16u;                 // per-lane byte offset
        const uint32_t ldsoff = ((uint32_t)(uintptr_t)sx) + voff;    // LDS byte address (low 32 bits of flat)
        asm volatile(
            "global_load_async_to_lds_b128 %0, %1, %2\n\t"
            "global_load_async_to_lds_b128 %0, %1, %2 offset:4096\n\t"
            "global_load_async_to_lds_b128 %0, %1, %2 offset:8192\n\t"
            "s_wait_asynccnt 0"
            :
            : "v"(ldsoff), "v"(voff), "s"(gbase)
            : "memory");
    }
    __syncthreads();

    float px = 0.f, py = 0.f, pz = 0.f;
    if (n < N) {
        if (fullBlock) {
            px = sx[tid * 3 + 0];   // stride-3 across lanes: bank-conflict free
            py = sx[tid * 3 + 1];
            pz = sx[tid * 3 + 2];
        } else {
            const float* p = xg + (size_t)n * 3;
            px = p[0]; py = p[1]; pz = p[2];
        }
    }
    if (n >= N) return;

    // clamp as in reference
    const float lo = 1e-6f, hi = 1.0f - 1e-6f;
    px = fminf(fmaxf(px, lo), hi);
    py = fminf(fmaxf(py, lo), hi);
    pz = fminf(fmaxf(pz, lo), hi);

    float acc[NLVL * 2];

#pragma unroll
    for (int l = 0; l < NLVL; ++l) {
        const v2f* __restrict__ tbl = (const v2f*)(grid + ((size_t)l << (LOG2T + 1)));
        const float r = kResM1[l];

        const float ux = px * r, uy = py * r, uz = pz * r;
        const float bx = floorf(ux), by = floorf(uy), bz = floorf(uz);
        const float tx = ux - bx,  ty = uy - by,  tz = uz - bz;

        const uint32_t X  = (uint32_t)bx;
        const uint32_t Y  = (uint32_t)by;
        const uint32_t Z  = (uint32_t)bz;
        const uint32_t X1 = X + 1u;

        // incremental spatial hash (uint32 wraparound matches jnp.uint32 exactly)
        const uint32_t hy0 = Y * PRIME_Y, hy1 = hy0 + PRIME_Y;
        const uint32_t hz0 = Z * PRIME_Z, hz1 = hz0 + PRIME_Z;
        const uint32_t e00 = hy0 ^ hz0, e01 = hy0 ^ hz1;
        const uint32_t e10 = hy1 ^ hz0, e11 = hy1 ^ hz1;

        // 8 irregular b64 gathers (L2-resident table)
        const v2f f000 = tbl[(X  ^ e00) & TMASK];
        const v2f f001 = tbl[(X  ^ e01) & TMASK];
        const v2f f010 = tbl[(X  ^ e10) & TMASK];
        const v2f f011 = tbl[(X  ^ e11) & TMASK];
        const v2f f100 = tbl[(X1 ^ e00) & TMASK];
        const v2f f101 = tbl[(X1 ^ e01) & TMASK];
        const v2f f110 = tbl[(X1 ^ e10) & TMASK];
        const v2f f111 = tbl[(X1 ^ e11) & TMASK];

        // trilinear weights, associated as (wx*wy)*wz to match jnp.prod order
        const float wx0 = 1.f - tx, wy0 = 1.f - ty, wz0 = 1.f - tz;
        const float a00 = wx0 * wy0, a01 = wx0 * ty, a10 = tx * wy0, a11 = tx * ty;
        const float w000 = a00 * wz0, w001 = a00 * tz;
        const float w010 = a01 * wz0, w011 = a01 * tz;
        const float w100 = a10 * wz0, w101 = a10 * tz;
        const float w110 = a11 * wz0, w111 = a11 * tz;

        float r0 = f000.x * w000;
        float r1 = f000.y * w000;
        r0 = fmaf(f001.x, w001, r0);  r1 = fmaf(f001.y, w001, r1);
        r0 = fmaf(f010.x, w010, r0);  r1 = fmaf(f010.y, w010, r1);
        r0 = fmaf(f011.x, w011, r0);  r1 = fmaf(f011.y, w011, r1);
        r0 = fmaf(f100.x, w100, r0);  r1 = fmaf(f100.y, w100, r1);
        r0 = fmaf(f101.x, w101, r0);  r1 = fmaf(f101.y, w101, r1);
        r0 = fmaf(f110.x, w110, r0);  r1 = fmaf(f110.y, w110, r1);
        r0 = fmaf(f111.x, w111, r0);  r1 = fmaf(f111.y, w111, r1);

        acc[2 * l + 0] = r0;
        acc[2 * l + 1] = r1;
    }

    // 128B contiguous per thread; non-temporal so the 64MB output stream
    // does not evict the L2-resident 64MB hash grid.
    v4f* op = (v4f*)(out + (size_t)n * (NLVL * 2));
#pragma unroll
    for (int k = 0; k < 8; ++k) {
        v4f v = { acc[4 * k + 0], acc[4 * k + 1], acc[4 * k + 2], acc[4 * k + 3] };
        __builtin_nontemporal_store(v, op + k);
    }
}

extern "C" void kernel_launch(void* const* d_in, const int* in_sizes, int n_in,
                              void* d_out, int out_size, void* d_ws, size_t ws_size,
                              hipStream_t stream)
{
    const float* x    = (const float*)d_in[0];   // [N,3] f32
    const float* grid = (const float*)d_in[1];   // [16, 2^19, 2] f32
    float*       out  = (float*)d_out;           // [N, 32] f32

    const int N      = in_sizes[0] / 3;
    const int blocks = (N + BLOCK - 1) / BLOCK;
    lotd_hashgrid_fwd<<<blocks, BLOCK, 0, stream>>>(x, grid, out, N);
}